// RWKV6_CrossAttention_62775241998772
// MI455X (gfx1250) — compile-verified
//
#include <hip/hip_runtime.h>
#include <math.h>

#define Hdim 256
#define NHEAD 4
#define DK 64
#define PLR 32
#define Bsz 2
#define LQ 8
#define LKs 512
#define NTOK (Bsz * LQ * LKs)   // 8192
#define EPSV 1e-5f

typedef float v2f __attribute__((ext_vector_type(2)));
typedef float v8f __attribute__((ext_vector_type(8)));

#define ACT_NONE  0
#define ACT_SILU  1
#define ACT_TANH  2
#define ACT_DECAY 3

// ---------------------------------------------------------------------------
// Phase 1a: per-token feature prep + low-rank mu networks.
// One block (256 threads) per token; GEMVs done from LDS.
// ---------------------------------------------------------------------------
__global__ void rwkv_features_kernel(
    const float* __restrict__ query, const float* __restrict__ keyval,
    const float* __restrict__ rwg_mu, const float* __restrict__ W_rwg0,
    const float* __restrict__ W_rwg2, const float* __restrict__ rwg_bias,
    const float* __restrict__ kv_mu, const float* __restrict__ W_kv0,
    const float* __restrict__ W_kv2, const float* __restrict__ kv_bias,
    float* __restrict__ xr, float* __restrict__ xw, float* __restrict__ xg,
    float* __restrict__ xk, float* __restrict__ xv)
{
    const int tid = blockIdx.x;          // token id: e*LK + l
    const int e   = tid >> 9;            // / 512
    const int l   = tid & 511;
    const int b   = e >> 3;              // / LQ
    const int t   = threadIdx.x;         // feature column 0..255

    __shared__ float xrwg_s[Hdim];
    __shared__ float xkv_s[Hdim];
    __shared__ float t1_s[3 * PLR];      // 96
    __shared__ float t2_s[2 * PLR];      // 64

    const float h  = keyval[((long)b * LKs + l) * Hdim + t];
    const float q  = query[(long)e * Hdim + t];
    const float hp = (l == 0) ? 0.0f : keyval[((long)b * LKs + (l - 1)) * Hdim + t];
    const float dc = q - h;
    const float d  = hp - h;

    xrwg_s[t] = h + dc * rwg_mu[t];
    xkv_s[t]  = h + d  * kv_mu[t];
    __syncthreads();

    if (t < 96) {
        float acc = 0.0f;
        #pragma unroll 8
        for (int i = 0; i < Hdim; ++i) acc += xrwg_s[i] * W_rwg0[i * 96 + t];
        t1_s[t] = tanhf(acc);
    } else if (t < 160) {
        const int j = t - 96;
        float acc = 0.0f;
        #pragma unroll 8
        for (int i = 0; i < Hdim; ++i) acc += xkv_s[i] * W_kv0[i * 64 + j];
        t2_s[j] = tanhf(acc);
    }
    __syncthreads();

    // rank-32 expansions back to H, per output column t
    float rmu = rwg_bias[t];
    float wmu = rwg_bias[Hdim + t];
    float gmu = rwg_bias[2 * Hdim + t];
    #pragma unroll
    for (int r = 0; r < PLR; ++r) {
        rmu += t1_s[r]            * W_rwg2[t * 96 + r];
        wmu += t1_s[PLR + r]      * W_rwg2[t * 96 + PLR + r];
        gmu += t1_s[2 * PLR + r]  * W_rwg2[t * 96 + 2 * PLR + r];
    }
    float kmu = kv_bias[t];
    float vmu = kv_bias[Hdim + t];
    #pragma unroll
    for (int r = 0; r < PLR; ++r) {
        kmu += t2_s[r]       * W_kv2[t * 64 + r];
        vmu += t2_s[PLR + r] * W_kv2[t * 64 + PLR + r];
    }

    const long o = (long)tid * Hdim + t;
    xr[o] = h + dc * rmu;
    xw[o] = h + dc * wmu;
    xg[o] = h + dc * gmu;
    xk[o] = h + d  * kmu;
    xv[o] = h + d  * vmu;
}

// ---------------------------------------------------------------------------
// WMMA fp32 GEMM: Y[M,N] = act(X[M,K] @ W[K,N] + bias[N]).
// 2D grid: blockIdx.x covers M in 128-row slabs (4 waves x 32 rows),
// blockIdx.y covers N in 32-col slabs. Each wave computes a 32x32 tile =
// 4 independent 16x16 accumulators via V_WMMA_F32_16X16X4_F32, so A/B loads
// are amortized 2x each and 4 chains keep the matrix pipe busy. Exact grid
// fit: no guards, EXEC all-ones throughout (WMMA requirement).
// A layout (16x4 f32): lane = m + 16*(k>=2), vgpr = k&1.
// B layout (4x16 f32): lane = n + 16*(k>=2), vgpr = k&1.
// C/D layout: vgpr r -> m = r + 8*(lane>>4), n = lane&15.
// ---------------------------------------------------------------------------
__device__ __forceinline__ float apply_act(float x, int act)
{
    switch (act) {
        case ACT_SILU:  return x / (1.0f + expf(-x));   // x * sigmoid(x)
        case ACT_TANH:  return tanhf(x);
        case ACT_DECAY: return expf(-expf(x));
        default:        return x;
    }
}

__global__ __launch_bounds__(128) void rwkv_wmma_gemm_kernel(
    const float* __restrict__ X, const float* __restrict__ W,
    const float* __restrict__ bias, float* __restrict__ Y,
    int K, int N, int act)
{
    const int wave  = threadIdx.x >> 5;
    const int lane  = threadIdx.x & 31;
    const int mbase = (blockIdx.x * 4 + wave) << 5;   // 32 rows per wave
    const int nbase = blockIdx.y << 5;                // 32 cols per wave

    const int sub  = lane & 15;                // m row (A) / n col (B)
    const int koff = (lane >> 4) << 1;         // 0 or 2

    const float* __restrict__ Arow0 = X + (long)(mbase + sub) * K + koff;
    const float* __restrict__ Arow1 = Arow0 + (long)16 * K;
    const float* __restrict__ Bcol0 = W + (long)koff * N + nbase + sub;
    const float* __restrict__ Bcol1 = Bcol0 + 16;

    v8f c00 = {}, c01 = {}, c10 = {}, c11 = {};
    #pragma unroll 4
    for (int k = 0; k < K; k += 4) {
        const v2f a0 = *(const v2f*)(Arow0 + k);       // global_load_b64
        const v2f a1 = *(const v2f*)(Arow1 + k);
        v2f b0, b1;
        b0.x = Bcol0[(long)k * N];
        b0.y = Bcol0[(long)(k + 1) * N];
        b1.x = Bcol1[(long)k * N];
        b1.y = Bcol1[(long)(k + 1) * N];
        c00 = __builtin_amdgcn_wmma_f32_16x16x4_f32(false, a0, false, b0, (short)0, c00, false, false);
        c01 = __builtin_amdgcn_wmma_f32_16x16x4_f32(false, a0, false, b1, (short)0, c01, false, false);
        c10 = __builtin_amdgcn_wmma_f32_16x16x4_f32(false, a1, false, b0, (short)0, c10, false, false);
        c11 = __builtin_amdgcn_wmma_f32_16x16x4_f32(false, a1, false, b1, (short)0, c11, false, false);
    }

    const int rbase = (lane >> 4) * 8;
    const int col0  = nbase + sub;
    const int col1  = col0 + 16;
    const float bv0 = bias ? bias[col0] : 0.0f;
    const float bv1 = bias ? bias[col1] : 0.0f;
    #pragma unroll
    for (int r = 0; r < 8; ++r) {
        const long row0 = (long)mbase + rbase + r;
        const long row1 = row0 + 16;
        Y[row0 * N + col0] = apply_act(c00[r] + bv0, act);
        Y[row0 * N + col1] = apply_act(c01[r] + bv1, act);
        Y[row1 * N + col0] = apply_act(c10[r] + bv0, act);
        Y[row1 * N + col1] = apply_act(c11[r] + bv1, act);
    }
}

// ---------------------------------------------------------------------------
// Phase 2: gated linear-attention scan. 64 blocks = (eb=16) x (NH=4) chains.
// 256 threads: thread t owns column v = t&63 and k-slice [16*(t>>6), +16).
// State S[k][v] in registers (16 floats/thread). o uses pre-update S.
// Next-step rows prefetched (global_prefetch_b8) to hide latency across the
// 512 serial iterations.
// ---------------------------------------------------------------------------
__global__ void rwkv_scan_kernel(
    const float* __restrict__ r, const float* __restrict__ k,
    const float* __restrict__ v, const float* __restrict__ decay,
    const float* __restrict__ u, float* __restrict__ o)
{
    const int e    = blockIdx.x >> 2;
    const int head = blockIdx.x & 3;
    const long base = ((long)e * LKs) * Hdim + head * DK;
    const int t    = threadIdx.x;
    const int vcol = t & 63;
    const int k0   = (t >> 6) * 16;

    float S[16];
    #pragma unroll
    for (int j = 0; j < 16; ++j) S[j] = 0.0f;

    __shared__ float rr[DK], kk[DK], vv[DK], dd[DK], uu[DK];
    __shared__ float part[256];

    if (t < DK) uu[t] = u[head * DK + t];
    __syncthreads();

    for (int step = 0; step < LKs; ++step) {
        const long p = base + (long)step * Hdim;
        if (t < 64) {
            rr[t] = r[p + t];
            __builtin_prefetch((const void*)(r + p + Hdim + t), 0, 1);
        } else if (t < 128) {
            kk[t - 64] = k[p + t - 64];
            __builtin_prefetch((const void*)(k + p + Hdim + t - 64), 0, 1);
        } else if (t < 192) {
            vv[t - 128] = v[p + t - 128];
            __builtin_prefetch((const void*)(v + p + Hdim + t - 128), 0, 1);
        } else {
            dd[t - 192] = decay[p + t - 192];
            __builtin_prefetch((const void*)(decay + p + Hdim + t - 192), 0, 1);
        }
        __syncthreads();

        const float vt = vv[vcol];
        float acc = 0.0f;
        #pragma unroll
        for (int j = 0; j < 16; ++j) {
            const int ki = k0 + j;
            const float kv = kk[ki] * vt;
            acc += rr[ki] * (S[j] + uu[ki] * kv);
            S[j] = dd[ki] * S[j] + kv;
        }
        part[t] = acc;
        __syncthreads();
        if (t < 64)
            o[p + t] = part[t] + part[t + 64] + part[t + 128] + part[t + 192];
        __syncthreads();
    }
}

// ---------------------------------------------------------------------------
// Phase 3a: per-head GroupNorm (4 groups of 64) * gn_w + gn_b, then * silu(g).
// One block (256 threads) per token.
// ---------------------------------------------------------------------------
__global__ void rwkv_gn_silu_kernel(
    const float* __restrict__ o, const float* __restrict__ gsilu,
    const float* __restrict__ gn_w, const float* __restrict__ gn_b,
    float* __restrict__ y)
{
    const long tok = blockIdx.x;
    const int t    = threadIdx.x;
    const int head = t >> 6;
    const int idx  = t & 63;

    const float val = o[tok * Hdim + t];
    __shared__ float red[256];

    red[t] = val;
    __syncthreads();
    for (int off = 32; off >= 1; off >>= 1) {
        if (idx < off) red[t] += red[t + off];
        __syncthreads();
    }
    const float mean = red[head * 64] * (1.0f / 64.0f);
    __syncthreads();

    const float dv = val - mean;
    red[t] = dv * dv;
    __syncthreads();
    for (int off = 32; off >= 1; off >>= 1) {
        if (idx < off) red[t] += red[t + off];
        __syncthreads();
    }
    const float var = red[head * 64] * (1.0f / 64.0f);

    const float ynorm = dv * rsqrtf(var + EPSV);
    y[tok * Hdim + t] = (ynorm * gn_w[t] + gn_b[t]) * gsilu[tok * Hdim + t];
}

// ---------------------------------------------------------------------------
extern "C" void kernel_launch(void* const* d_in, const int* in_sizes, int n_in,
                              void* d_out, int out_size, void* d_ws, size_t ws_size,
                              hipStream_t stream)
{
    const float* query    = (const float*)d_in[0];
    const float* keyval   = (const float*)d_in[1];
    const float* rwg_mu   = (const float*)d_in[2];
    const float* W_rwg0   = (const float*)d_in[3];
    const float* W_rwg2   = (const float*)d_in[4];
    const float* rwg_bias = (const float*)d_in[5];
    const float* kv_mu    = (const float*)d_in[6];
    const float* W_kv0    = (const float*)d_in[7];
    const float* W_kv2    = (const float*)d_in[8];
    const float* kv_bias  = (const float*)d_in[9];
    const float* Wr       = (const float*)d_in[10];
    const float* Wa       = (const float*)d_in[11];
    const float* Wb       = (const float*)d_in[12];
    const float* bb       = (const float*)d_in[13];
    const float* Wk       = (const float*)d_in[14];
    const float* Wv       = (const float*)d_in[15];
    const float* Wg       = (const float*)d_in[16];
    const float* u        = (const float*)d_in[17];
    const float* gn_w     = (const float*)d_in[18];
    const float* gn_b     = (const float*)d_in[19];
    const float* Wo       = (const float*)d_in[20];

    float* ws = (float*)d_ws;
    const long SZ = (long)NTOK * Hdim;     // 2,097,152 floats per buffer
    float* xr    = ws + 0 * SZ;
    float* xw    = ws + 1 * SZ;
    float* xg    = ws + 2 * SZ;
    float* xk    = ws + 3 * SZ;
    float* xv    = ws + 4 * SZ;
    float* rbuf  = ws + 5 * SZ;
    float* kbuf  = ws + 6 * SZ;
    float* vbuf  = ws + 7 * SZ;
    float* dbuf  = ws + 8 * SZ;            // decay
    float* gsbuf = ws + 9 * SZ;            // silu(g)
    float* twbuf = ws + 10 * SZ;           // NTOK*64 = SZ/4
    float* obuf  = xr;                     // reuse after r-GEMM consumed xr
    float* ybuf  = xw;                     // reuse after w-path consumed xw

    rwkv_features_kernel<<<NTOK, 256, 0, stream>>>(
        query, keyval, rwg_mu, W_rwg0, W_rwg2, rwg_bias,
        kv_mu, W_kv0, W_kv2, kv_bias, xr, xw, xg, xk, xv);

    auto gemm = [&](const float* X, const float* W, const float* bias,
                    float* Y, int M, int K, int N, int act) {
        // M multiple of 128 (8192), N multiple of 32 (256 or 64): exact fit.
        dim3 grid(M / 128, N / 32);
        rwkv_wmma_gemm_kernel<<<grid, 128, 0, stream>>>(X, W, bias, Y, K, N, act);
    };

    gemm(xr, Wr, nullptr, rbuf,  NTOK, 256, 256, ACT_NONE);
    gemm(xk, Wk, nullptr, kbuf,  NTOK, 256, 256, ACT_NONE);
    gemm(xv, Wv, nullptr, vbuf,  NTOK, 256, 256, ACT_NONE);
    gemm(xg, Wg, nullptr, gsbuf, NTOK, 256, 256, ACT_SILU);
    gemm(xw, Wa, nullptr, twbuf, NTOK, 256, 64,  ACT_TANH);
    gemm(twbuf, Wb, bb, dbuf,    NTOK, 64,  256, ACT_DECAY);

    rwkv_scan_kernel<<<Bsz * LQ * NHEAD, 256, 0, stream>>>(rbuf, kbuf, vbuf, dbuf, u, obuf);

    rwkv_gn_silu_kernel<<<NTOK, 256, 0, stream>>>(obuf, gsbuf, gn_w, gn_b, ybuf);

    gemm(ybuf, Wo, nullptr, (float*)d_out, NTOK, 256, 256, ACT_NONE);
}